// SafeRocketLeagueGCN_30588757082542
// MI455X (gfx1250) — compile-verified
//
#include <hip/hip_runtime.h>
#include <math.h>

typedef __attribute__((ext_vector_type(2))) float v2f;
typedef __attribute__((ext_vector_type(8))) float v8f;

#define HID 32

// ---------------------------------------------------------------- utilities
__global__ void k_fill(float* __restrict__ p, float v, int n) {
  int i = blockIdx.x * blockDim.x + threadIdx.x;
  if (i < n) p[i] = v;
}

// deg[dst] += w  (deg pre-initialized to 1.0 == self-loop weight)
__global__ void k_deg(const int* __restrict__ dst, const float* __restrict__ w,
                      float* __restrict__ deg, int E) {
  int e = blockIdx.x * blockDim.x + threadIdx.x;
  if (e < E) unsafeAtomicAdd(&deg[dst[e]], w[e]);
}

// deg -> dinv = rsqrt(deg) (deg >= 1 always, but keep reference's guard)
__global__ void k_rsqrt(float* __restrict__ d, int n) {
  int i = blockIdx.x * blockDim.x + threadIdx.x;
  if (i < n) { float v = d[i]; d[i] = v > 0.f ? rsqrtf(v) : 0.f; }
}

// ------------------------------------------------------------- WMMA GEMM 1
// h[N,32] = x[N,4] @ W1[4,32] : one wave per 16-row tile, 2 WMMAs (N-tiles)
__global__ void k_gemm1(const float* __restrict__ x, const float* __restrict__ W,
                        float* __restrict__ h, int tiles) {
  int wave = (blockIdx.x * blockDim.x + threadIdx.x) >> 5;
  int lane = threadIdx.x & 31;
  if (wave >= tiles) return;              // wave-uniform: EXEC stays all-ones
  int row0 = wave * 16;
  int half = lane >> 4;                   // K-half: lanes 0-15 -> K{0,1}, 16-31 -> K{2,3}
  int l    = lane & 15;
  int k0   = half * 2;
  // A tile: row M=l, two consecutive K values
  const float* xr = x + (size_t)(row0 + l) * 4 + k0;
  v2f a; a.x = xr[0]; a.y = xr[1];
#pragma unroll
  for (int nt = 0; nt < 2; ++nt) {
    int col = nt * 16 + l;                // B column N=l (+16 for second tile)
    v2f b; b.x = W[(k0 + 0) * HID + col]; b.y = W[(k0 + 1) * HID + col];
    v8f c = {};
    c = __builtin_amdgcn_wmma_f32_16x16x4_f32(false, a, false, b,
                                              (short)0, c, false, false);
#pragma unroll
    for (int v = 0; v < 8; ++v)           // D: M = v + 8*half, N = col
      h[(size_t)(row0 + v + 8 * half) * HID + col] = c[v];
  }
}

// scalar tail (only launched if N % 16 != 0)
__global__ void k_gemm1_tail(const float* __restrict__ x, const float* __restrict__ W,
                             float* __restrict__ h, int row_start, int N) {
  int idx = blockIdx.x * blockDim.x + threadIdx.x;
  int r = row_start + (idx >> 5), j = idx & 31;
  if (r >= N) return;
  float acc = 0.f;
#pragma unroll
  for (int k = 0; k < 4; ++k) acc += x[(size_t)r * 4 + k] * W[k * HID + j];
  h[(size_t)r * HID + j] = acc;
}

// ------------------------------------------------------------- WMMA GEMM 2
// h[N,32] = z[N,32] @ W2[32,32] : 8 chained K=4 WMMA slices per N-tile
__global__ void k_gemm2(const float* __restrict__ z, const float* __restrict__ W,
                        float* __restrict__ h, int tiles) {
  int wave = (blockIdx.x * blockDim.x + threadIdx.x) >> 5;
  int lane = threadIdx.x & 31;
  if (wave >= tiles) return;
  int row0 = wave * 16;
  int half = lane >> 4;
  int l    = lane & 15;
  const float* zr = z + (size_t)(row0 + l) * HID;
#pragma unroll
  for (int nt = 0; nt < 2; ++nt) {
    int col = nt * 16 + l;
    v8f c = {};
#pragma unroll
    for (int kk = 0; kk < 8; ++kk) {      // K = kk*4 .. kk*4+3
      int k0 = kk * 4 + half * 2;
      v2f a; a.x = zr[k0]; a.y = zr[k0 + 1];
      v2f b; b.x = W[(k0 + 0) * HID + col]; b.y = W[(k0 + 1) * HID + col];
      c = __builtin_amdgcn_wmma_f32_16x16x4_f32(false, a, false, b,
                                                (short)0, c, false, false);
    }
#pragma unroll
    for (int v = 0; v < 8; ++v)
      h[(size_t)(row0 + v + 8 * half) * HID + col] = c[v];
  }
}

__global__ void k_gemm2_tail(const float* __restrict__ z, const float* __restrict__ W,
                             float* __restrict__ h, int row_start, int N) {
  int idx = blockIdx.x * blockDim.x + threadIdx.x;
  int r = row_start + (idx >> 5), j = idx & 31;
  if (r >= N) return;
  float acc = 0.f;
  for (int k = 0; k < HID; ++k) acc += z[(size_t)r * HID + k] * W[k * HID + j];
  h[(size_t)r * HID + j] = acc;
}

// ------------------------------------------------------------ aggregation
// agg[i,:] = h[i,:] * dinv[i]^2  (self-loop contribution, weight 1)
__global__ void k_selfloop(const float* __restrict__ h, const float* __restrict__ dinv,
                           float* __restrict__ agg, int N) {
  int idx = blockIdx.x * blockDim.x + threadIdx.x;
  if (idx >= N * HID) return;
  float di = dinv[idx >> 5];
  agg[idx] = h[idx] * di * di;
}

// one wave per edge: lane j does agg[dst][j] += h[src][j]*dinv[src]*w*dinv[dst]
__global__ void k_edge(const int* __restrict__ src, const int* __restrict__ dst,
                       const float* __restrict__ w, const float* __restrict__ dinv,
                       const float* __restrict__ h, float* __restrict__ agg, int E) {
  unsigned tid = blockIdx.x * blockDim.x + threadIdx.x;
  int e = tid >> 5, lane = tid & 31;
  if (e >= E) return;
  int s = src[e], d = dst[e];
  float coef = dinv[s] * w[e] * dinv[d];  // broadcast loads, one req/wave
  unsafeAtomicAdd(&agg[(size_t)d * HID + lane], h[(size_t)s * HID + lane] * coef);
}

__global__ void k_bias_relu(float* __restrict__ a, const float* __restrict__ b, int total) {
  int idx = blockIdx.x * blockDim.x + threadIdx.x;
  if (idx < total) a[idx] = fmaxf(a[idx] + b[idx & (HID - 1)], 0.f);
}

// ---------------------------------------------------------------- pooling
__global__ void k_pool(const float* __restrict__ z, const int* __restrict__ batch,
                       float* __restrict__ sums, float* __restrict__ cnt, int N) {
  unsigned tid = blockIdx.x * blockDim.x + threadIdx.x;
  int i = tid >> 5, lane = tid & 31;
  if (i >= N) return;
  int g = batch[i];
  unsafeAtomicAdd(&sums[(size_t)g * HID + lane], z[(size_t)i * HID + lane]);
  if (lane == 0) unsafeAtomicAdd(&cnt[g], 1.0f);
}

__global__ void k_heads(const float* __restrict__ sums, const float* __restrict__ cnt,
                        const float* __restrict__ Wo, const float* __restrict__ bo,
                        const float* __restrict__ Wb, const float* __restrict__ bb,
                        float* __restrict__ out, int G) {
  int g = blockIdx.x * blockDim.x + threadIdx.x;
  if (g >= G) return;
  float inv = 1.0f / fmaxf(cnt[g], 1.0f);
  float so = 0.f, sb = 0.f;
#pragma unroll
  for (int j = 0; j < HID; ++j) {
    float p = sums[(size_t)g * HID + j] * inv;
    so += p * Wo[j];
    sb += p * Wb[j];
  }
  out[g]     = 1.0f / (1.0f + expf(-(so + bo[0])));
  out[G + g] = 1.0f / (1.0f + expf(-(sb + bb[0])));
}

// ------------------------------------------------------------------ driver
extern "C" void kernel_launch(void* const* d_in, const int* in_sizes, int n_in,
                              void* d_out, int out_size, void* d_ws, size_t ws_size,
                              hipStream_t stream) {
  const float* x     = (const float*)d_in[0];
  const int*   ei    = (const int*)  d_in[1];
  const float* ew    = (const float*)d_in[2];
  const int*   batch = (const int*)  d_in[3];
  const float* W1    = (const float*)d_in[5];
  const float* b1    = (const float*)d_in[6];
  const float* W2    = (const float*)d_in[7];
  const float* b2    = (const float*)d_in[8];
  const float* Wo    = (const float*)d_in[9];
  const float* bo    = (const float*)d_in[10];
  const float* Wb    = (const float*)d_in[11];
  const float* bb    = (const float*)d_in[12];

  const int N = in_sizes[0] / 4;   // 100000
  const int E = in_sizes[2];       // 1600000
  const int G = out_size / 2;      // 16384
  const int* src = ei;             // edge_index[0]
  const int* dst = ei + E;         // edge_index[1]

  float* ws   = (float*)d_ws;
  float* dinv = ws;                                  // N
  float* bufA = dinv + N;                            // N*32  (h1, later h2)
  float* bufB = bufA + (size_t)N * HID;              // N*32  (agg1/z1, later agg2/z2)
  float* sums = bufB + (size_t)N * HID;              // G*32
  float* cnt  = sums + (size_t)G * HID;              // G
  float* out  = (float*)d_out;

  const int B = 256;
  auto nb = [](long n, int b) { return (int)((n + b - 1) / b); };

  // init: deg = 1 (self-loop weight), sums = 0, cnt = 0
  k_fill<<<nb(N, B), B, 0, stream>>>(dinv, 1.0f, N);
  k_fill<<<nb((long)G * HID, B), B, 0, stream>>>(sums, 0.f, G * HID);
  k_fill<<<nb(G, B), B, 0, stream>>>(cnt, 0.f, G);

  // weighted degree -> dinv
  k_deg<<<nb(E, B), B, 0, stream>>>(dst, ew, dinv, E);
  k_rsqrt<<<nb(N, B), B, 0, stream>>>(dinv, N);

  const int tiles = N / 16, rem = N % 16;

  // ---- layer 1: h1 = x@W1 ; agg = scatter(norm*h1) ; z1 = relu(agg+b1)
  k_gemm1<<<nb((long)tiles * 32, B), B, 0, stream>>>(x, W1, bufA, tiles);
  if (rem)
    k_gemm1_tail<<<nb((long)rem * 32, B), B, 0, stream>>>(x, W1, bufA, tiles * 16, N);
  k_selfloop<<<nb((long)N * HID, B), B, 0, stream>>>(bufA, dinv, bufB, N);
  k_edge<<<nb((long)E * 32, B), B, 0, stream>>>(src, dst, ew, dinv, bufA, bufB, E);
  k_bias_relu<<<nb((long)N * HID, B), B, 0, stream>>>(bufB, b1, N * HID);

  // ---- layer 2: h2 = z1@W2 ; agg ; z2 = relu(agg+b2)
  k_gemm2<<<nb((long)tiles * 32, B), B, 0, stream>>>(bufB, W2, bufA, tiles);
  if (rem)
    k_gemm2_tail<<<nb((long)rem * 32, B), B, 0, stream>>>(bufB, W2, bufA, tiles * 16, N);
  k_selfloop<<<nb((long)N * HID, B), B, 0, stream>>>(bufA, dinv, bufB, N);
  k_edge<<<nb((long)E * 32, B), B, 0, stream>>>(src, dst, ew, dinv, bufA, bufB, E);
  k_bias_relu<<<nb((long)N * HID, B), B, 0, stream>>>(bufB, b2, N * HID);

  // ---- pooling + heads
  k_pool<<<nb((long)N * 32, B), B, 0, stream>>>(bufB, batch, sums, cnt, N);
  k_heads<<<nb(G, B), B, 0, stream>>>(sums, cnt, Wo, bo, Wb, bb, out, G);
}